// SAGEEncoder_19851338842495
// MI455X (gfx1250) — compile-verified
//
#include <hip/hip_runtime.h>

// GraphSAGE 2-layer encoder for MI455X (gfx1250, wave32).
// Phases per layer: scatter-mean (L2-resident f32 atomics) then fused
// GEMM (V_WMMA_F32_16X16X4_F32) + bias + ReLU.

typedef __attribute__((ext_vector_type(2))) float v2f;
typedef __attribute__((ext_vector_type(8))) float v8f;

// ---------------------------------------------------------------- zero fill
__global__ void zero_f32(float* __restrict__ p, long n) {
  long i = (long)blockIdx.x * blockDim.x + threadIdx.x;
  long stride = (long)gridDim.x * blockDim.x;
  for (; i < n; i += stride) p[i] = 0.0f;
}

// ---------------------------------------------------------------- degrees
__global__ void degree_kernel(const int* __restrict__ dst,
                              float* __restrict__ deg, int E) {
  int e = blockIdx.x * blockDim.x + threadIdx.x;
  if (e < E) atomicAdd(&deg[dst[e]], 1.0f);
}

// ---------------------------------------------------------------- scatter-add
// One wave32 per edge; each lane moves 4 floats (float4 gather + 4 atomics).
// hs: element stride of h (1 = dense [N,128]; 2 = interleaved [N,128,2] slot0).
__global__ void scatter_add(const float* __restrict__ h, int hs,
                            const int* __restrict__ src,
                            const int* __restrict__ dst,
                            float* __restrict__ agg, int E) {
  int t = blockIdx.x * blockDim.x + threadIdx.x;
  int e = t >> 5;
  int lane = t & 31;
  if (e >= E) return;
  int s = src[e];
  int d = dst[e];
  float* p = agg + (long)d * 128 + lane * 4;
  if (hs == 1) {
    const float4 v = *(const float4*)(h + (long)s * 128 + lane * 4);
    atomicAdd(p + 0, v.x);
    atomicAdd(p + 1, v.y);
    atomicAdd(p + 2, v.z);
    atomicAdd(p + 3, v.w);
  } else {
    const float* hp = h + ((long)s * 128 + lane * 4) * hs;
    atomicAdd(p + 0, hp[0 * hs]);
    atomicAdd(p + 1, hp[1 * hs]);
    atomicAdd(p + 2, hp[2 * hs]);
    atomicAdd(p + 3, hp[3 * hs]);
  }
}

// ---------------------------------------------------------------- fused GEMM
// out_row = relu( (agg/deg) @ Wl^T + b + h @ Wr^T )
// Block = 256 threads = 8 waves; block handles 16 rows, wave w handles output
// columns [16w,16w+16). WMMA f32 16x16x4, K=128 => 32 steps x 2 matmuls.
__global__ void __launch_bounds__(256) sage_gemm_relu(
    const float* __restrict__ h, int hs,
    const float* __restrict__ agg,
    const float* __restrict__ deg,
    const float* __restrict__ Wl,     // [128,128] row-major
    const float* __restrict__ Wr,     // [128,128] row-major
    const float* __restrict__ bias,   // [128]
    float* __restrict__ hout,         // dense [N,128] next features, or null
    float* __restrict__ out,          // [N,128,2]
    int slot) {
  __shared__ float sA[16][132];  // mean rows (padded to dodge bank conflicts)
  __shared__ float sH[16][132];  // h rows
  __shared__ float sD[16];       // 1/max(deg,1)

  const int t = threadIdx.x;
  const int row0 = blockIdx.x * 16;

  if (t < 16) {
    float dg = deg[row0 + t];
    sD[t] = 1.0f / fmaxf(dg, 1.0f);
  }
  __syncthreads();

#pragma unroll
  for (int i = 0; i < 8; ++i) {
    int idx = i * 256 + t;  // coalesced across the block
    int r = idx >> 7;
    int c = idx & 127;
    sA[r][c] = agg[(long)(row0 + r) * 128 + c] * sD[r];
    sH[r][c] = h[((long)(row0 + r) * 128 + c) * hs];
  }
  __syncthreads();

  const int lane = t & 31;   // wave32
  const int wave = t >> 5;
  const int m = lane & 15;   // A-row within tile / B-col within tile
  const int hi = lane >> 4;  // selects K pair {2hi, 2hi+1}
  const int col0 = wave * 16;
  const int wrow = col0 + m;  // weight row == output column

  const float bv = bias[wrow];
  v8f acc = {bv, bv, bv, bv, bv, bv, bv, bv};

  const float* wlrow = Wl + (long)wrow * 128;
  const float* wrrow = Wr + (long)wrow * 128;

#pragma unroll 4
  for (int s = 0; s < 32; ++s) {
    const int k = 4 * s + 2 * hi;
    // A fragment (16x4 f32): lane m holds row m, K = k..k+1
    v2f a1 = *(const v2f*)&sA[m][k];
    // B fragment (4x16 f32): B[k][j] = Wl[j][k]
    v2f b1 = *(const v2f*)(wlrow + k);
    acc = __builtin_amdgcn_wmma_f32_16x16x4_f32(false, a1, false, b1,
                                                (short)0, acc, false, false);
    v2f a2 = *(const v2f*)&sH[m][k];
    v2f b2 = *(const v2f*)(wrrow + k);
    acc = __builtin_amdgcn_wmma_f32_16x16x4_f32(false, a2, false, b2,
                                                (short)0, acc, false, false);
  }

  // C/D layout: acc[r] = C[M = r + 8*hi][N = m]
  const int col = col0 + m;
#pragma unroll
  for (int r = 0; r < 8; ++r) {
    float v = fmaxf(acc[r], 0.0f);
    long o = (long)(row0 + 8 * hi + r) * 128 + col;
    if (hout) hout[o] = v;
    out[o * 2 + slot] = v;
  }
}

// ---------------------------------------------------------------- launcher
extern "C" void kernel_launch(void* const* d_in, const int* in_sizes, int n_in,
                              void* d_out, int out_size, void* d_ws,
                              size_t ws_size, hipStream_t stream) {
  (void)n_in;
  (void)out_size;
  const float* x = (const float*)d_in[0];
  const int* ei = (const int*)d_in[1];
  const float* W1l = (const float*)d_in[2];
  const float* b1 = (const float*)d_in[3];
  const float* W1r = (const float*)d_in[4];
  const float* W2l = (const float*)d_in[5];
  const float* b2 = (const float*)d_in[6];
  const float* W2r = (const float*)d_in[7];
  float* out = (float*)d_out;

  const int N = in_sizes[0] / 128;  // 100000 (divisible by 16)
  const int E = in_sizes[1] / 2;    // 1600000
  const int* src = ei;
  const int* dst = ei + E;

  // workspace: deg[N] | agg[N*128] | x1[N*128]
  float* deg = (float*)d_ws;
  size_t degN = ((size_t)N + 255) & ~(size_t)255;
  float* agg = deg + degN;
  float* x1 = agg + (size_t)N * 128;
  size_t need = (degN + (size_t)N * 128 * 2) * sizeof(float);
  bool dense_x1 = (ws_size >= need);

  const int ZB = 2048, ZT = 256;
  int scatter_blocks = (int)(((long)E * 32 + 255) / 256);
  int gemm_blocks = N / 16;

  // ---- layer 1
  zero_f32<<<ZB, ZT, 0, stream>>>(deg, (long)degN + (long)N * 128);
  degree_kernel<<<(E + 255) / 256, 256, 0, stream>>>(dst, deg, E);
  scatter_add<<<scatter_blocks, 256, 0, stream>>>(x, 1, src, dst, agg, E);
  sage_gemm_relu<<<gemm_blocks, 256, 0, stream>>>(
      x, 1, agg, deg, W1l, W1r, b1, dense_x1 ? x1 : nullptr, out, 0);

  // ---- layer 2
  const float* h2 = dense_x1 ? (const float*)x1 : (const float*)out;
  int hs2 = dense_x1 ? 1 : 2;  // fallback reads x1 from out slot 0 (stride 2)
  zero_f32<<<ZB, ZT, 0, stream>>>(agg, (long)N * 128);
  scatter_add<<<scatter_blocks, 256, 0, stream>>>(h2, hs2, src, dst, agg, E);
  sage_gemm_relu<<<gemm_blocks, 256, 0, stream>>>(
      h2, hs2, agg, deg, W2l, W2r, b2, nullptr, out, 1);
}